// GEctPointsBULayer_44255343018851
// MI455X (gfx1250) — compile-verified
//
#include <hip/hip_runtime.h>

typedef __attribute__((ext_vector_type(2))) float v2f;
typedef __attribute__((ext_vector_type(8))) float v8f;

#define R_CONST       1.1f
#define SCALE_CONST   100.0f
#define BUMP_STEPS    32
#define NUM_THETAS    32
#define NUM_FEATURES  128
#define LOG2E_F       1.4426950408889634f

// sigma(z) = 0.5 + 0.5*tanh(z/2): one TRANS op (v_tanh_f32) instead of two
// (v_exp_f32 + v_rcp_f32). TRANS throughput is the bottleneck here.
__device__ __forceinline__ float fast_sigmoid(float z) {
#if __has_builtin(__builtin_amdgcn_tanhf)
    return __builtin_fmaf(0.5f, __builtin_amdgcn_tanhf(0.5f * z), 0.5f);
#else
    float e = __builtin_amdgcn_exp2f(-z * LOG2E_F);
    return __builtin_amdgcn_rcpf(1.0f + e);
#endif
}

// One block = 256 threads = 8 waves; each wave owns one 16-node strip.
// Per wave: nh(16 nodes x 32 thetas) via 64x V_WMMA_F32_16X16X4_F32, then
// 32 filtration steps of sigmoids accumulated into per-block LDS tiles.
// Tiles are split by half-wave (hi) so ds_add_f32 never collides within a wave.
__global__ __launch_bounds__(256)
void ect_fused_kernel(const float* __restrict__ x,
                      const int*   __restrict__ batch,
                      const float* __restrict__ v,
                      float* __restrict__ out,
                      int n_nodes)
{
    __shared__ float vsh[NUM_THETAS * NUM_FEATURES];          // 16 KB, v[t][f]
    __shared__ float tile[2 * 2 * BUMP_STEPS * NUM_THETAS];   // 16 KB: [slot][hi][s][t]
    __shared__ int   s_g0, s_g1;

    const int tid  = threadIdx.x;
    const int wave = tid >> 5;
    const int lane = tid & 31;

    // Stage v into LDS (coalesced float4 copy: 4096 floats / 256 threads)
    {
        const float4* vsrc = (const float4*)v;
        float4*       vdst = (float4*)vsh;
        for (int i = tid; i < (NUM_THETAS * NUM_FEATURES) / 4; i += 256)
            vdst[i] = vsrc[i];
    }
    for (int i = tid; i < 2 * 2 * BUMP_STEPS * NUM_THETAS; i += 256)
        tile[i] = 0.0f;

    if (tid == 0) {
        int base = blockIdx.x * 128;
        int last = base + 127; if (last >= n_nodes) last = n_nodes - 1;
        s_g0 = batch[base];
        s_g1 = batch[last];
    }
    __syncthreads();

    const int g0 = s_g0, g1 = s_g1;
    const int strip = blockIdx.x * 8 + wave;       // 16-node strip id (wave-uniform)

    if (strip * 16 < n_nodes) {                    // wave-uniform guard: EXEC all-1s for WMMA
        const int hi = lane >> 4;                  // half-wave select
        const int lm = lane & 15;
        const int node_base = strip * 16;

        // Per-lane base pointers per the 32-bit WMMA A/B fragment layouts
        const float* arow  = x   + (size_t)(node_base + lm) * NUM_FEATURES + 2 * hi;
        const float* b0row = vsh + (size_t)lm        * NUM_FEATURES + 2 * hi;
        const float* b1row = vsh + (size_t)(lm + 16) * NUM_FEATURES + 2 * hi;

        v8f acc0 = {};   // nh[node rows][theta 0..15]
        v8f acc1 = {};   // nh[node rows][theta 16..31]

        #pragma unroll 8
        for (int k = 0; k < NUM_FEATURES / 4; ++k) {
            const int kb = k * 4;
            if ((kb & 31) == 0)
                __builtin_prefetch(arow + kb + 64, 0, 3);   // WGP-scope prefetch, next 256B
            v2f a  = *(const v2f*)(arow  + kb);
            v2f b0 = *(const v2f*)(b0row + kb);
            v2f b1 = *(const v2f*)(b1row + kb);
            acc0 = __builtin_amdgcn_wmma_f32_16x16x4_f32(false, a, false, b0,
                                                         (short)0, acc0, false, false);
            acc1 = __builtin_amdgcn_wmma_f32_16x16x4_f32(false, a, false, b1,
                                                         (short)0, acc1, false, false);
        }

        // Graph labels of this lane's 8 node rows (sorted => endpoint check suffices)
        int grow[8];
        #pragma unroll
        for (int r = 0; r < 8; ++r)
            grow[r] = batch[node_base + 8 * hi + r];
        const bool gsame = (grow[0] == grow[7]);
        const int t0 = lm, t1 = lm + 16;
        float* tile0 = tile + hi * (BUMP_STEPS * NUM_THETAS);          // slot 0, this hi
        float* tile1 = tile0 + 2 * (BUMP_STEPS * NUM_THETAS);          // slot 1, this hi

        const float step = 2.0f * R_CONST / (float)(BUMP_STEPS - 1);

        #pragma unroll 1
        for (int s = 0; s < BUMP_STEPS; ++s) {
            const float lin_s  = -R_CONST + (float)s * step;
            const float base_s = fast_sigmoid(SCALE_CONST * (lin_s - R_CONST));
            const int so = s * NUM_THETAS;

            if (gsame) {
                // Pre-sum 8 rows in registers -> one LDS atomic per (s, theta)
                float sum0 = 0.f, sum1 = 0.f;
                #pragma unroll
                for (int r = 0; r < 8; ++r) {
                    sum0 += fast_sigmoid(SCALE_CONST * (lin_s - acc0[r]));
                    sum1 += fast_sigmoid(SCALE_CONST * (lin_s - acc1[r]));
                }
                sum0 -= 8.0f * base_s;
                sum1 -= 8.0f * base_s;
                const int g = grow[0];
                if (g == g0) {
                    atomicAdd(&tile0[so + t0], sum0);
                    atomicAdd(&tile0[so + t1], sum1);
                } else if (g == g1) {
                    atomicAdd(&tile1[so + t0], sum0);
                    atomicAdd(&tile1[so + t1], sum1);
                } else {
                    atomicAdd(&out[(size_t)g * 1024 + so + t0], sum0);
                    atomicAdd(&out[(size_t)g * 1024 + so + t1], sum1);
                }
            } else {
                // Rare: graph boundary inside this lane's 8 rows
                #pragma unroll
                for (int r = 0; r < 8; ++r) {
                    float e0 = fast_sigmoid(SCALE_CONST * (lin_s - acc0[r])) - base_s;
                    float e1 = fast_sigmoid(SCALE_CONST * (lin_s - acc1[r])) - base_s;
                    const int g = grow[r];
                    if (g == g0) {
                        atomicAdd(&tile0[so + t0], e0);
                        atomicAdd(&tile0[so + t1], e1);
                    } else if (g == g1) {
                        atomicAdd(&tile1[so + t0], e0);
                        atomicAdd(&tile1[so + t1], e1);
                    } else {
                        atomicAdd(&out[(size_t)g * 1024 + so + t0], e0);
                        atomicAdd(&out[(size_t)g * 1024 + so + t1], e1);
                    }
                }
            }
        }
    }
    __syncthreads();

    // Flush block tiles to global, merging the two hi halves.
    // out layout [g][s][t] matches tile [s][t] flat.
    for (int i = tid; i < 1024; i += 256)
        atomicAdd(&out[(size_t)g0 * 1024 + i], tile[i] + tile[1024 + i]);
    if (g1 != g0)
        for (int i = tid; i < 1024; i += 256)
            atomicAdd(&out[(size_t)g1 * 1024 + i], tile[2048 + i] + tile[3072 + i]);
}

extern "C" void kernel_launch(void* const* d_in, const int* in_sizes, int n_in,
                              void* d_out, int out_size, void* d_ws, size_t ws_size,
                              hipStream_t stream) {
    // setup_inputs order: x [N,128] f32, batch [N] int, num_graphs scalar, v [32,128] f32
    const float* x     = (const float*)d_in[0];
    const int*   batch = (const int*)d_in[1];
    const float* v     = (const float*)d_in[3];
    const int n_nodes  = in_sizes[0] / NUM_FEATURES;

    (void)hipMemsetAsync(d_out, 0, (size_t)out_size * sizeof(float), stream);

    const int strips = (n_nodes + 15) / 16;
    const int blocks = (strips + 7) / 8;       // 8 waves (strips) per block
    ect_fused_kernel<<<blocks, 256, 0, stream>>>(x, batch, v, (float*)d_out, n_nodes);
}